// DigitCaps_40106404610625
// MI455X (gfx1250) — compile-verified
//
#include <hip/hip_runtime.h>

// DigitCaps dynamic routing for MI455X (gfx1250, wave32).
// Strategy: recompute u_hat from L2-resident x/W each routing pass using
// V_WMMA_F32_16X16X4_F32; keep routing state as a running sum of v's
// (b_ij^(t) = u_hat . sum_{tau<t} v_tau), so no 189MB u_hat tensor and no
// 12MB b_ij tensor ever touch memory.

#define B_   256
#define I_   1152
#define DI_  8
#define O_   10
#define DO_  16
#define OD_  160        // O_*DO_
#define NCHUNK 16       // i-chunks
#define IC   72         // I_/NCHUNK
#define NW   4          // waves per workgroup
#define WG_THREADS 128  // NW * 32 (wave32)

typedef float v2f __attribute__((ext_vector_type(2)));
typedef float v8f __attribute__((ext_vector_type(8)));

__global__ __launch_bounds__(256) void caps_zero_kernel(float* p, int n) {
    int idx = blockIdx.x * blockDim.x + threadIdx.x;
    if (idx < n) p[idx] = 0.0f;
}

// One workgroup per (16-batch tile, 72-i chunk). Each wave owns a subset of
// i's; per i it builds the [16 x 160] u_hat tile with f32 WMMA, then lane
// pairs do the routing softmax + weighted accumulation from LDS.
__global__ __launch_bounds__(WG_THREADS) void caps_iter_kernel(
    const float* __restrict__ x,     // [B_, I_, DI_]
    const float* __restrict__ W,     // [I_, O_, DO_, DI_]
    const float* __restrict__ vacc,  // [B_, OD_]  (sum of previous v's)
    float* __restrict__ s_part)      // [NCHUNK, B_, OD_] partial s
{
    __shared__ float u_tile[NW][16][OD_];  // per-wave u_hat tile
    __shared__ float s_acc [NW][16][OD_];  // per-wave s partial
    __shared__ float vtile [16][OD_];      // staged v-accumulator for 16 rows

    const int btile = blockIdx.x;
    const int ich   = blockIdx.y;
    const int b0    = btile * 16;
    const int tid   = threadIdx.x;
    const int w     = tid >> 5;
    const int lane  = tid & 31;

    // stage vacc tile; zero per-wave s accumulators
    for (int idx = tid; idx < 16 * OD_; idx += WG_THREADS) {
        int r = idx / OD_, od = idx % OD_;
        vtile[r][od] = vacc[(size_t)(b0 + r) * OD_ + od];
        #pragma unroll
        for (int ww = 0; ww < NW; ++ww) s_acc[ww][r][od] = 0.0f;
    }
    __syncthreads();

    // WMMA lane roles (16x16x4 f32):
    //  A 16x4: lanes 0-15 -> K=0(v0),1(v1); lanes 16-31 -> K=2(v0),3(v1)
    //  B 4x16: lane = N column, same K split across lane halves
    //  C 16x16: lane = N column, VGPR r -> M = r + 8*(lane>=16)
    const int rowA  = lane & 15;
    const int khalf = (lane >> 4) << 1;   // 0 or 2
    const int ncol  = lane & 15;
    const int chalf = lane >> 4;

    for (int ii = w; ii < IC; ii += NW) {   // uniform trip count: 18 per wave
        const int i = ich * IC + ii;

        // A: x[b0+rowA, i, 0..7] split into two 16x4 operands (K 0-3, 4-7)
        const float* xp = x + ((size_t)(b0 + rowA) * I_ + i) * DI_;
        v2f a0 = *(const v2f*)(xp + khalf);
        v2f a1 = *(const v2f*)(xp + khalf + 4);

        // per output capsule: C_o[16b x 16d] = x_tile(16x8) * W[i,o]^T(8x16)
        for (int o = 0; o < O_; ++o) {
            const float* wp = W + (((size_t)i * O_ + o) * DO_ + ncol) * DI_;
            v2f bm0 = *(const v2f*)(wp + khalf);
            v2f bm1 = *(const v2f*)(wp + khalf + 4);
            v8f acc = {0.f, 0.f, 0.f, 0.f, 0.f, 0.f, 0.f, 0.f};
            acc = __builtin_amdgcn_wmma_f32_16x16x4_f32(
                      false, a0, false, bm0, (short)0, acc, false, false);
            acc = __builtin_amdgcn_wmma_f32_16x16x4_f32(
                      false, a1, false, bm1, (short)0, acc, false, false);
            #pragma unroll
            for (int r = 0; r < 8; ++r)
                u_tile[w][r + 8 * chalf][o * DO_ + ncol] = acc[r];
        }
        __syncthreads();

        // routing: lane pair (2 lanes) per batch row, 8 d's each
        const int row = lane >> 1;
        const int db  = (lane & 1) * 8;
        float bv[O_];
        #pragma unroll
        for (int o = 0; o < O_; ++o) {
            float p = 0.0f;
            #pragma unroll
            for (int j = 0; j < 8; ++j)
                p += u_tile[w][row][o * DO_ + db + j] *
                     vtile[row][o * DO_ + db + j];
            p += __shfl_xor(p, 1, 32);   // combine the two d-halves
            bv[o] = p;                   // b_ij (zero vacc -> uniform softmax)
        }
        float m = bv[0];
        #pragma unroll
        for (int o = 1; o < O_; ++o) m = fmaxf(m, bv[o]);
        float sum = 0.0f;
        #pragma unroll
        for (int o = 0; o < O_; ++o) { bv[o] = __expf(bv[o] - m); sum += bv[o]; }
        const float inv = 1.0f / sum;
        #pragma unroll
        for (int o = 0; o < O_; ++o) {
            const float c = bv[o] * inv;
            #pragma unroll
            for (int j = 0; j < 8; ++j)
                s_acc[w][row][o * DO_ + db + j] +=
                    c * u_tile[w][row][o * DO_ + db + j];
        }
        __syncthreads();
    }

    // flush: reduce the NW per-wave partials, store chunk partial to scratch
    for (int idx = tid; idx < 16 * OD_; idx += WG_THREADS) {
        int r = idx / OD_, od = idx % OD_;
        float v = 0.0f;
        #pragma unroll
        for (int ww = 0; ww < NW; ++ww) v += s_acc[ww][r][od];
        s_part[((size_t)ich * B_ + (b0 + r)) * OD_ + od] = v;
    }
}

// One block per batch element: reduce chunk partials, squash, update the
// v-accumulator (or emit final output).
__global__ __launch_bounds__(OD_) void caps_squash_kernel(
    const float* __restrict__ s_part,  // [NCHUNK, B_, OD_]
    float* __restrict__ vacc,          // [B_, OD_]
    float* __restrict__ out,           // [B_, OD_] (final iter only)
    int final_iter)
{
    __shared__ float sv[OD_];
    const int b  = blockIdx.x;
    const int od = threadIdx.x;

    float sel = 0.0f;
    #pragma unroll
    for (int ch = 0; ch < NCHUNK; ++ch)
        sel += s_part[((size_t)ch * B_ + b) * OD_ + od];
    sv[od] = sel;
    __syncthreads();

    const int o = od >> 4;
    float sq = 0.0f;
    #pragma unroll
    for (int d = 0; d < DO_; ++d) { float u = sv[o * DO_ + d]; sq += u * u; }
    const float v = sel * sq / ((1.0f + sq) * sqrtf(sq));

    if (final_iter) out[(size_t)b * OD_ + od] = v;
    else            vacc[(size_t)b * OD_ + od] += v;
}

extern "C" void kernel_launch(void* const* d_in, const int* in_sizes, int n_in,
                              void* d_out, int out_size, void* d_ws, size_t ws_size,
                              hipStream_t stream) {
    (void)in_sizes; (void)n_in; (void)out_size; (void)ws_size;
    const float* x = (const float*)d_in[0];   // [256,1152,8]
    const float* W = (const float*)d_in[1];   // [1152,10,16,8]
    float* out     = (float*)d_out;           // [256,10,16]

    float* vacc   = (float*)d_ws;                    // 40960 floats
    float* s_part = vacc + (size_t)B_ * OD_;         // 16*40960 floats (~2.6MB)

    caps_zero_kernel<<<(B_ * OD_ + 255) / 256, 256, 0, stream>>>(vacc, B_ * OD_);

    dim3 grid(B_ / 16, NCHUNK);
    for (int it = 0; it < 3; ++it) {
        caps_iter_kernel<<<grid, WG_THREADS, 0, stream>>>(x, W, vacc, s_part);
        caps_squash_kernel<<<B_, OD_, 0, stream>>>(s_part, vacc, out, it == 2 ? 1 : 0);
    }
}